// CapsuleLayer_11862699671991
// MI455X (gfx1250) — compile-verified
//
#include <hip/hip_runtime.h>
#include <hip/hip_bf16.h>

typedef float v2f __attribute__((ext_vector_type(2)));
typedef float v8f __attribute__((ext_vector_type(8)));

#define B_ 128
#define U_ 8
#define C_ 2048
#define J_ 32
#define S_ 16
#define K_ (C_ * U_)  // 16384

// ---------------------------------------------------------------------------
// One-time layout transforms (per launch; deterministic).
// WT2: [J][K/2][S][2] with k=(c*8+u):  WT2[((j*8192+kp)*16+s)*2+t] = W[c][j][s][u]
// -> A-fragment (A[s,k], M=s, K-pair contiguous) is one coalesced b64 per lane.
// ---------------------------------------------------------------------------
__global__ void transpose_W_kernel(const float* __restrict__ W, float* __restrict__ WT2) {
    int o = blockIdx.x * 256 + threadIdx.x;      // < 8388608
    int t  = o & 1;
    int s  = (o >> 1) & 15;
    int kp = (o >> 5) & 8191;
    int j  = o >> 18;
    int k  = kp * 2 + t;
    int c  = k >> 3, u = k & 7;
    WT2[o] = W[(((size_t)c * J_ + j) * S_ + s) * U_ + u];
}

// xT2: [K/2][B][2]:  xT2[(kp*128+b)*2+t] = x[b][u][c], k=kp*2+t
__global__ void transpose_x_kernel(const float* __restrict__ x, float* __restrict__ xT2) {
    int o = blockIdx.x * 256 + threadIdx.x;      // < 2097152
    int t  = o & 1;
    int b  = (o >> 1) & 127;
    int kp = o >> 8;
    int k  = kp * 2 + t;
    int c  = k >> 3, u = k & 7;
    xT2[o] = x[((size_t)b * U_ + u) * C_ + c];
}

// ---------------------------------------------------------------------------
// Softmax over C (axis=0) for each j column of b[C,J].
// ---------------------------------------------------------------------------
__global__ void softmax_c_kernel(const float* __restrict__ bb, float* __restrict__ cc) {
    int j = blockIdx.x;
    int tid = threadIdx.x;                        // 256 threads
    __shared__ float red[256];
    float m = -1e30f;
    for (int c = tid; c < C_; c += 256) m = fmaxf(m, bb[c * J_ + j]);
    red[tid] = m; __syncthreads();
    for (int off = 128; off > 0; off >>= 1) {
        if (tid < off) red[tid] = fmaxf(red[tid], red[tid + off]);
        __syncthreads();
    }
    float mx = red[0]; __syncthreads();
    float sum = 0.f;
    for (int c = tid; c < C_; c += 256) sum += __expf(bb[c * J_ + j] - mx);
    red[tid] = sum; __syncthreads();
    for (int off = 128; off > 0; off >>= 1) {
        if (tid < off) red[tid] += red[tid + off];
        __syncthreads();
    }
    float inv = 1.0f / red[0];
    for (int c = tid; c < C_; c += 256) cc[c * J_ + j] = __expf(bb[c * J_ + j] - mx) * inv;
}

// ---------------------------------------------------------------------------
// s[b,j,s] = sum_{c,u} c[c,j] * W[c,j,s,u] * x[b,u,c]
// 2x2 register blocking: each wave computes (2 j's) x (2 b-tiles) = 4 WMMA
// accumulators from 2 A-frags + 2 B-frags per K-step (halves L2 traffic/WMMA).
// Grid: jp(16) x btp(4) x ks(4) = 256 blocks; 8 waves split each K-slice.
// Block writes its exclusive spart[ks] slice -> deterministic (no atomics).
// ---------------------------------------------------------------------------
__global__ __launch_bounds__(256) void s_kernel(
    const float* __restrict__ WT2, const float* __restrict__ xT2,
    const float* __restrict__ cc, float* __restrict__ spart) {
    int bx  = blockIdx.x;
    int ks  = bx & 3;             // K-split slice
    int btp = (bx >> 2) & 3;      // b-tile pair
    int jp  = bx >> 4;            // j pair
    int j0  = jp * 2;
    int bt0 = btp * 2;

    int tid  = threadIdx.x;
    int w    = tid >> 5;
    int lane = tid & 31;
    int perm = (lane >> 4) * 2;   // K sub-offset per ISA A/B fragment layout
    int ln   = lane & 15;

    const float* wt0 = WT2 + (size_t)j0 * K_ * S_;
    const float* wt1 = wt0 + (size_t)K_ * S_;

    v8f a00 = {}, a01 = {}, a10 = {}, a11 = {};
    int k0 = ks * 4096 + w * 512;
    for (int kb = k0; kb < k0 + 512; kb += 4) {
        int c = kb >> 3;                              // 4 k's share one c
        v2f cw = *(const v2f*)(cc + (size_t)c * J_ + j0);  // wave-uniform pair
        int kk = kb + perm;                           // even
        size_t kpi = (size_t)(kk >> 1);
        v2f fa0 = *(const v2f*)(wt0 + (kpi * 16 + ln) * 2);
        v2f fa1 = *(const v2f*)(wt1 + (kpi * 16 + ln) * 2);
        v2f fb0 = *(const v2f*)(xT2 + (kpi * 128 + bt0 * 16 + ln) * 2);
        v2f fb1 = *(const v2f*)(xT2 + (kpi * 128 + (bt0 + 1) * 16 + ln) * 2);
        fa0 = fa0 * cw.x;
        fa1 = fa1 * cw.y;
        a00 = __builtin_amdgcn_wmma_f32_16x16x4_f32(false, fa0, false, fb0, (short)0, a00, false, false);
        a01 = __builtin_amdgcn_wmma_f32_16x16x4_f32(false, fa0, false, fb1, (short)0, a01, false, false);
        a10 = __builtin_amdgcn_wmma_f32_16x16x4_f32(false, fa1, false, fb0, (short)0, a10, false, false);
        a11 = __builtin_amdgcn_wmma_f32_16x16x4_f32(false, fa1, false, fb1, (short)0, a11, false, false);
    }

    // reduce the 8 wave-partials of 4 tiles through LDS (fixed order)
    __shared__ float red[8][4][256];
#pragma unroll
    for (int r = 0; r < 8; r++) {
        int m = r + 8 * (lane >> 4);                  // C/D layout: M = r + 8*(lane/16)
        red[w][0][m * 16 + ln] = a00[r];
        red[w][1][m * 16 + ln] = a01[r];
        red[w][2][m * 16 + ln] = a10[r];
        red[w][3][m * 16 + ln] = a11[r];
    }
    __syncthreads();
    int m = tid >> 4, n = tid & 15;                   // m = s index, n = b offset
    float* sp = spart + (size_t)ks * (B_ * J_ * S_);
#pragma unroll
    for (int t = 0; t < 4; t++) {
        int jj = t >> 1, bb = t & 1;
        float sum = 0.f;
#pragma unroll
        for (int ww = 0; ww < 8; ww++) sum += red[ww][t][tid];
        sp[(size_t)((bt0 + bb) * 16 + n) * (J_ * S_) + (j0 + jj) * S_ + m] = sum;
    }
}

// ---------------------------------------------------------------------------
// Squash: sum the 4 K-split partials (fixed order), norm over axis=1 (J).
// ---------------------------------------------------------------------------
__global__ void squash_kernel(const float* __restrict__ spart, float* __restrict__ vbuf,
                              float* __restrict__ out, int write_out) {
    int b = blockIdx.x;
    int tid = threadIdx.x;                      // 512 = J*S, tid = j*16+s
    __shared__ float red[512];
    size_t idx = (size_t)b * 512 + tid;
    float val = spart[idx] + spart[idx + 65536] + spart[idx + 2 * 65536] + spart[idx + 3 * 65536];
    red[tid] = val * val;
    __syncthreads();
    for (int jo = 256; jo >= 16; jo >>= 1) {    // reduce over J per s
        if (tid < jo) red[tid] += red[tid + jo];
        __syncthreads();
    }
    float msq = red[tid & 15];
    float scale = msq / ((1.0f + msq) * sqrtf(msq));
    float vv = val * scale;
    vbuf[idx] = vv;
    if (write_out) out[idx] = vv;
}

// ---------------------------------------------------------------------------
// b[c,j] += (1/B) * sum_{b,s,u} W[c,j,s,u] x[b,u,c] v[b,j,s]
// Block = (c-tile of 16, j-quad); wave w owns j = jq*4+w and all 8 u's.
// v[b,j,s] staged once into LDS (8KB/wave); x A-frags shared via L0 across
// the 4 waves. Exclusive (c,j) ownership -> plain RMW on b, deterministic.
// ---------------------------------------------------------------------------
__global__ __launch_bounds__(128) void bupd_kernel(
    const float* __restrict__ x, const float* __restrict__ W,
    const float* __restrict__ vbuf, float* __restrict__ bb) {
    int ct = blockIdx.x;          // 0..127
    int jq = blockIdx.y;          // 0..7
    int cbase = ct * 16;
    int tid  = threadIdx.x;       // 128 = 4 waves
    int w    = tid >> 5;
    int lane = tid & 31;
    int perm = (lane >> 4) * 2;
    int ln   = lane & 15;
    int j    = jq * 4 + w;        // this wave's output capsule

    __shared__ float vlds[4][B_ * S_];           // [wave][b*16+s], 32KB
    for (int i = lane; i < B_ * S_; i += 32)
        vlds[w][i] = vbuf[(size_t)(i >> 4) * (J_ * S_) + j * S_ + (i & 15)];
    __syncthreads();

    float sacc[8];
#pragma unroll
    for (int r = 0; r < 8; r++) sacc[r] = 0.f;

    for (int u = 0; u < U_; u++) {
        v8f acc = {};
        for (int kb = 0; kb < B_; kb += 4) {     // K = batch
            int k = kb + perm;
            v2f a, bf;
            a.x  = x[((size_t)k * U_ + u) * C_ + cbase + ln];
            a.y  = x[((size_t)(k + 1) * U_ + u) * C_ + cbase + ln];
            bf.x = vlds[w][k * 16 + ln];
            bf.y = vlds[w][(k + 1) * 16 + ln];
            acc = __builtin_amdgcn_wmma_f32_16x16x4_f32(
                false, a, false, bf, (short)0, acc, false, false);
        }
#pragma unroll
        for (int r = 0; r < 8; r++) {
            int c = cbase + r + 8 * (lane >> 4);
            float wv = W[((size_t)c * J_ + j) * (S_ * U_) + ln * U_ + u];
            float p = acc[r] * wv;               // acc element is (c, s=ln)
            p += __shfl_xor(p, 1);               // reduce over s within 16-lane half
            p += __shfl_xor(p, 2);
            p += __shfl_xor(p, 4);
            p += __shfl_xor(p, 8);
            sacc[r] += p;
        }
    }
    if (ln == 0) {                               // lanes 0 (c'=r) and 16 (c'=r+8)
        int h = lane >> 4;
#pragma unroll
        for (int r = 0; r < 8; r++) {
            size_t bi = (size_t)(cbase + r + 8 * h) * J_ + j;
            bb[bi] += sacc[r] * (1.0f / 128.0f);
        }
    }
}

// ---------------------------------------------------------------------------
extern "C" void kernel_launch(void* const* d_in, const int* in_sizes, int n_in,
                              void* d_out, int out_size, void* d_ws, size_t ws_size,
                              hipStream_t stream) {
    (void)in_sizes; (void)n_in; (void)out_size; (void)ws_size;
    const float* x = (const float*)d_in[0];   // [128,8,2048]
    const float* W = (const float*)d_in[1];   // [1,2048,32,16,8]
    float* out = (float*)d_out;               // [128,32,16,1]

    float* ws    = (float*)d_ws;
    float* WT2   = ws;                        // 8,388,608 floats
    float* xT2   = WT2 + 8388608;             // 2,097,152 floats
    float* bb    = xT2 + 2097152;             // 65,536
    float* cc    = bb + 65536;                // 65,536
    float* spart = cc + 65536;                // 4 * 65,536 (K-split partials)
    float* vbuf  = spart + 4 * 65536;         // 65,536

    hipMemsetAsync(bb, 0, 65536 * sizeof(float), stream);
    transpose_W_kernel<<<8388608 / 256, 256, 0, stream>>>(W, WT2);
    transpose_x_kernel<<<2097152 / 256, 256, 0, stream>>>(x, xT2);

    for (int it = 0; it < 3; it++) {
        softmax_c_kernel<<<32, 256, 0, stream>>>(bb, cc);
        s_kernel<<<256, 256, 0, stream>>>(WT2, xT2, cc, spart);
        squash_kernel<<<128, 512, 0, stream>>>(spart, vbuf, out, it == 2 ? 1 : 0);
        if (it < 2) {
            bupd_kernel<<<dim3(128, 8), 128, 0, stream>>>(x, W, vbuf, bb);
        }
    }
}